// SkillLibrary_91087666414187
// MI455X (gfx1250) — compile-verified
//
#include <hip/hip_runtime.h>
#include <hip/hip_bf16.h>

typedef __attribute__((ext_vector_type(2))) float v2f;
typedef __attribute__((ext_vector_type(8))) float v8f;

#define B_ 8
#define L_ 4096
#define D_ 2048
#define E_ 256
#define N_ 32
#define R_ 16
#define EPS 1e-6f
#define LCH 8               // l-chunks for deterministic 2-stage pooling
#define MPAD 16             // M padded to WMMA tile height
#define KC (N_ * R_)        // 512: contraction dim of the up-projection
#define OUT_ELEMS (B_ * L_ * D_)

// ---- workspace layout (floats) ----
#define WS_PARTIAL   0                          // [B][LCH][D]      = 131072
#define WS_POOLED    (WS_PARTIAL + B_*LCH*D_)   // [16][D] zero-pad = 32768
#define WS_QRAW      (WS_POOLED + MPAD*D_)      // [B][E]           = 2048
#define WS_Q         (WS_QRAW + B_*E_)          // [B][E]           = 2048
#define WS_WEIGHTS   (WS_Q + B_*E_)             // [B][N]           = 256
#define WS_COEF      (WS_WEIGHTS + B_*N_)       // [16][512] 0-pad  = 8192
#define WS_SKILL     (WS_COEF + MPAD*KC)        // [B][D]           = 16384

// ---------------- K1a: partial sums of x over L-chunks (deterministic) -------
__global__ void k_pool_partial(const float* __restrict__ x, float* __restrict__ ws) {
    int blk = blockIdx.x;              // B * (D/256) * LCH = 512 blocks
    int b  = blk >> 6;
    int rem = blk & 63;
    int dc = rem >> 3;
    int lc = rem & 7;
    int d  = dc * 256 + threadIdx.x;
    const float* xp = x + ((size_t)b * L_ + (size_t)lc * (L_ / LCH)) * D_ + d;
    float s = 0.f;
    for (int l = 0; l < L_ / LCH; ++l) s += xp[(size_t)l * D_];
    ws[WS_PARTIAL + ((size_t)b * LCH + lc) * D_ + d] = s;
}

// ---------------- K1b: reduce partials -> pooled mean (rows 8..15 zeroed) ----
__global__ void k_pool_reduce(float* __restrict__ ws) {
    int idx = blockIdx.x * 256 + threadIdx.x;   // 128 blocks -> 32768 (16 rows)
    int b = idx / D_, d = idx % D_;
    float v = 0.f;
    if (b < B_) {
        float s = 0.f;
        for (int lc = 0; lc < LCH; ++lc) s += ws[WS_PARTIAL + ((size_t)b * LCH + lc) * D_ + d];
        v = s * (1.0f / (float)L_);
    }
    ws[WS_POOLED + (size_t)b * D_ + d] = v;
}

// ---------------- K2: qraw = pooled @ Wq^T + bq  via WMMA f32 16x16x4 --------
// one block, 16 waves; wave w owns E-tile [16w, 16w+16). No divergence: pooled
// is zero-padded to 16 rows so every lane loads unconditionally (b64 pairs).
__global__ void k_qproj_wmma(const float* __restrict__ Wq, const float* __restrict__ bq,
                             float* __restrict__ ws) {
    int lane = threadIdx.x & 31;
    int wave = threadIdx.x >> 5;
    int nbase = wave * 16;
    int m = lane & 15;
    int kh = (lane >> 4) << 1;       // 0 or 2 (K-half select)
    int n = lane & 15;
    const float* pooled = ws + WS_POOLED;
    v8f acc = {};
    #pragma unroll 4
    for (int k = 0; k < D_; k += 4) {
        v2f a = *(const v2f*)(pooled + (size_t)m * D_ + k + kh);
        v2f bfr = *(const v2f*)(Wq + (size_t)(nbase + n) * D_ + k + kh);
        acc = __builtin_amdgcn_wmma_f32_16x16x4_f32(false, a, false, bfr,
                                                    (short)0, acc, false, false);
    }
    int half = lane >> 4;
    #pragma unroll
    for (int j = 0; j < 8; ++j) {
        int mm = j + 8 * half;
        if (mm < B_) ws[WS_QRAW + (size_t)mm * E_ + nbase + n] = acc[j] + bq[nbase + n];
    }
}

// ---------------- K3: rmsnorm(q) + scores + softmax -> weights ---------------
__global__ void k_gate(const float* __restrict__ gq, const float* __restrict__ emb,
                       const int* __restrict__ active_mask,
                       float* __restrict__ ws, float* __restrict__ out_weights) {
    __shared__ float red[256];
    __shared__ float sc[256];
    int t = threadIdx.x;
    // rmsnorm per batch row over E=256
    for (int b = 0; b < B_; ++b) {
        float v = ws[WS_QRAW + (size_t)b * E_ + t];
        red[t] = v * v;
        __syncthreads();
        for (int s = 128; s > 0; s >>= 1) {
            if (t < s) red[t] += red[t + s];
            __syncthreads();
        }
        float scale = rsqrtf(red[0] / (float)E_ + EPS);
        __syncthreads();
        ws[WS_Q + (size_t)b * E_ + t] = v * scale * gq[t];
    }
    __syncthreads();
    // scores: thread t -> (b = t/32, n = t%32)
    int b = t >> 5, n = t & 31;
    float s = 0.f;
    const float* qb = ws + WS_Q + (size_t)b * E_;
    const float* en = emb + (size_t)n * E_;
    for (int e = 0; e < E_; ++e) s += qb[e] * en[e];
    float mk = (float)active_mask[n];
    s = s * 0.0625f - (1.0f - mk) * 1e9f;   // /sqrt(256) and mask bias
    sc[t] = s;
    __syncthreads();
    // softmax per batch (8 threads, one per b)
    if (t < B_) {
        float mx = -1e30f;
        for (int i = 0; i < N_; ++i) mx = fmaxf(mx, sc[t * N_ + i]);
        float sum = 0.f;
        for (int i = 0; i < N_; ++i) sum += __expf(sc[t * N_ + i] - mx);
        float inv = 1.0f / sum;
        for (int i = 0; i < N_; ++i) {
            float w = __expf(sc[t * N_ + i] - mx) * inv;
            ws[WS_WEIGHTS + t * N_ + i] = w;
            out_weights[t * N_ + i] = w;
        }
    }
}

// ---------------- K4: per-slot t = pooled @ A_k, scaled -> coef  (WMMA) ------
// grid 32 blocks (one slot each), 1 wave per block. Writes all 16 coef rows
// (pad rows get 0) so K5 can load unconditionally.
__global__ void k_lora_down_wmma(const float* __restrict__ A, const int* __restrict__ active_mask,
                                 float* __restrict__ ws) {
    int kslot = blockIdx.x;
    int lane = threadIdx.x & 31;
    int m = lane & 15;
    int kh = (lane >> 4) << 1;
    int r = lane & 15;
    const float* pooled = ws + WS_POOLED;
    const float* Ak = A + (size_t)kslot * D_ * R_;    // [D][R]
    v8f acc = {};
    #pragma unroll 4
    for (int k = 0; k < D_; k += 4) {
        v2f a = *(const v2f*)(pooled + (size_t)m * D_ + k + kh);
        v2f bfr;
        bfr.x = Ak[(size_t)(k + kh) * R_ + r];
        bfr.y = Ak[(size_t)(k + kh + 1) * R_ + r];
        acc = __builtin_amdgcn_wmma_f32_16x16x4_f32(false, a, false, bfr,
                                                    (short)0, acc, false, false);
    }
    int half = lane >> 4;
    float mk = (float)active_mask[kslot];
    #pragma unroll
    for (int j = 0; j < 8; ++j) {
        int mm = j + 8 * half;
        float w = (mm < B_) ? ws[WS_WEIGHTS + mm * N_ + kslot] : 0.f;
        ws[WS_COEF + (size_t)mm * KC + kslot * R_ + r] =
            acc[j] * w * mk * (1.0f / (float)R_);
    }
}

// ---------------- K5: skill_out = coef @ Bm_flat  (WMMA) ---------------------
// grid 128 blocks (D/16 tiles), 1 wave per block; K = N*R = 512
__global__ void k_lora_up_wmma(const float* __restrict__ Bm, float* __restrict__ ws) {
    int nbase = blockIdx.x * 16;
    int lane = threadIdx.x & 31;
    int m = lane & 15;
    int kh = (lane >> 4) << 1;
    int n = lane & 15;
    const float* coef = ws + WS_COEF;                 // [16][512], zero-padded
    v8f acc = {};
    #pragma unroll 4
    for (int k = 0; k < KC; k += 4) {
        v2f a = *(const v2f*)(coef + (size_t)m * KC + k + kh);
        v2f bfr;
        bfr.x = Bm[(size_t)(k + kh) * D_ + nbase + n];
        bfr.y = Bm[(size_t)(k + kh + 1) * D_ + nbase + n];
        acc = __builtin_amdgcn_wmma_f32_16x16x4_f32(false, a, false, bfr,
                                                    (short)0, acc, false, false);
    }
    int half = lane >> 4;
    #pragma unroll
    for (int j = 0; j < 8; ++j) {
        int mm = j + 8 * half;
        if (mm < B_) ws[WS_SKILL + (size_t)mm * D_ + nbase + n] = acc[j];
    }
}

// ---------------- K6: fused residual + RMSNorm over D, one row per block -----
__global__ void k_final_norm(const float* __restrict__ x, const float* __restrict__ gnorm,
                             const float* __restrict__ ws, float* __restrict__ out) {
    __shared__ float red[256];
    int row = blockIdx.x;             // 32768 rows
    int b = row >> 12;                // / 4096
    int t = threadIdx.x;
    const float* xrow = x + (size_t)row * D_;
    const float* so = ws + WS_SKILL + (size_t)b * D_;
    float y[8];
    float ss = 0.f;
    int d0 = t * 8;
    #pragma unroll
    for (int i = 0; i < 8; i += 4) {
        float4 xv = *(const float4*)(xrow + d0 + i);
        float4 sv = *(const float4*)(so + d0 + i);
        y[i + 0] = xv.x + 0.1f * sv.x;
        y[i + 1] = xv.y + 0.1f * sv.y;
        y[i + 2] = xv.z + 0.1f * sv.z;
        y[i + 3] = xv.w + 0.1f * sv.w;
        ss += y[i] * y[i] + y[i + 1] * y[i + 1] + y[i + 2] * y[i + 2] + y[i + 3] * y[i + 3];
    }
    red[t] = ss;
    __syncthreads();
    for (int s = 128; s > 0; s >>= 1) {
        if (t < s) red[t] += red[t + s];
        __syncthreads();
    }
    float scale = rsqrtf(red[0] / (float)D_ + EPS);
    float* orow = out + (size_t)row * D_;
    #pragma unroll
    for (int i = 0; i < 8; i += 4) {
        float4 ov;
        ov.x = y[i + 0] * scale * gnorm[d0 + i + 0];
        ov.y = y[i + 1] * scale * gnorm[d0 + i + 1];
        ov.z = y[i + 2] * scale * gnorm[d0 + i + 2];
        ov.w = y[i + 3] * scale * gnorm[d0 + i + 3];
        *(float4*)(orow + d0 + i) = ov;
    }
}

extern "C" void kernel_launch(void* const* d_in, const int* in_sizes, int n_in,
                              void* d_out, int out_size, void* d_ws, size_t ws_size,
                              hipStream_t stream) {
    const float* x      = (const float*)d_in[0];
    const float* Wq     = (const float*)d_in[1];
    const float* bq     = (const float*)d_in[2];
    const float* gq     = (const float*)d_in[3];
    const float* emb    = (const float*)d_in[4];
    const float* A      = (const float*)d_in[5];
    const float* Bm     = (const float*)d_in[6];
    const float* gnorm  = (const float*)d_in[7];
    const int*   amask  = (const int*)d_in[8];
    float* out = (float*)d_out;
    float* ws  = (float*)d_ws;

    k_pool_partial<<<B_ * (D_ / 256) * LCH, 256, 0, stream>>>(x, ws);
    k_pool_reduce<<<(MPAD * D_) / 256, 256, 0, stream>>>(ws);
    k_qproj_wmma<<<1, 512, 0, stream>>>(Wq, bq, ws);
    k_gate<<<1, 256, 0, stream>>>(gq, emb, amask, ws, out + (size_t)OUT_ELEMS);
    k_lora_down_wmma<<<N_, 32, 0, stream>>>(A, amask, ws);
    k_lora_up_wmma<<<D_ / 16, 32, 0, stream>>>(Bm, ws);
    k_final_norm<<<B_ * L_, 256, 0, stream>>>(x, gnorm, ws, out);
}